// idct_8x8_40114994545195
// MI455X (gfx1250) — compile-verified
//
#include <hip/hip_runtime.h>

typedef float v2f __attribute__((ext_vector_type(2)));
typedef float v8f __attribute__((ext_vector_type(8)));

#if defined(__HIP_DEVICE_COMPILE__) && !__has_builtin(__builtin_amdgcn_wmma_f32_16x16x4_f32)
#error "DEVICE PASS: __builtin_amdgcn_wmma_f32_16x16x4_f32 not declared for gfx1250"
#endif

// D[n,c] = sum_k A[n,k] * B'[k,c] + 128, A = image (N x 64), B' = 0.25*alpha[k]*basis[k,c]
// Each wave32 computes a 16x64 output tile via 64 x V_WMMA_F32_16X16X4_F32.
__global__ __launch_bounds__(256) void idct8x8_wmma_f32_kernel(
    const float* __restrict__ image,   // [N, 64]
    const float* __restrict__ alpha,   // [64]
    const float* __restrict__ basis,   // [64, 64]  (k = x*8+y, c = u*8+v)
    float* __restrict__ out,           // [N, 64]
    int n_tiles)                       // N / 16
{
#if defined(__HIP_DEVICE_COMPILE__)
    __shared__ float Bs[64 * 64];      // 16 KB: scaled basis, shared by all waves

    // Build B'[k][c] = 0.25 * alpha[k] * basis[k][c] cooperatively.
    for (int i = threadIdx.x; i < 64 * 64; i += 256) {
        Bs[i] = 0.25f * alpha[i >> 6] * basis[i];
    }
    __syncthreads();

    const int lane = threadIdx.x & 31;
    const int wave = threadIdx.x >> 5;          // 0..7
    const int n    = lane & 15;                 // column / row-within-tile index
    const int half = lane >> 4;                 // 0: lanes 0-15, 1: lanes 16-31
    const int gwaves = gridDim.x * 8;

    for (int t = blockIdx.x * 8 + wave; t < n_tiles; t += gwaves) {
        const float* At = image + (size_t)t * (16 * 64);
        float*       Ot = out   + (size_t)t * (16 * 64);

        // --- A fragments, one v2f per K-chunk of 4 ---------------------------
        // ISA 16x4 fp32 A layout: VGPR0 = {K=0 | K=2}, VGPR1 = {K=1 | K=3}
        // => lane (n, half) holds A[row=n][4*kc + 2*half + {0,1}]
        v2f a[16];
        const float* arow = At + n * 64 + 2 * half;
        #pragma unroll
        for (int kc = 0; kc < 16; ++kc)
            a[kc] = *(const v2f*)(arow + kc * 4);

        // --- accumulators, +128 bias folded into C ---------------------------
        v8f acc[4];
        #pragma unroll
        for (int nt = 0; nt < 4; ++nt)
            #pragma unroll
            for (int r = 0; r < 8; ++r)
                acc[nt][r] = 128.0f;

        // --- 16 K-chunks x 4 N-tiles = 64 WMMAs ------------------------------
        #pragma unroll
        for (int kc = 0; kc < 16; ++kc) {
            const int krow = kc * 4 + 2 * half;     // B rows this lane supplies
            const float* b0 = &Bs[krow * 64 + n];
            #pragma unroll
            for (int nt = 0; nt < 4; ++nt) {
                v2f b;
                b.x = b0[nt * 16];          // B'[krow  ][16*nt + n]
                b.y = b0[64 + nt * 16];     // B'[krow+1][16*nt + n]
                acc[nt] = __builtin_amdgcn_wmma_f32_16x16x4_f32(
                    /*neg_a=*/false, a[kc],
                    /*neg_b=*/false, b,
                    /*c_mod=*/(short)0, acc[nt],
                    /*reuse_a=*/false, /*reuse_b=*/false);
            }
        }

        // --- store D per the 16x16 f32 C/D layout ----------------------------
        // VGPR r: lanes 0-15 -> row r, lanes 16-31 -> row r+8; col = 16*nt + n
        #pragma unroll
        for (int nt = 0; nt < 4; ++nt)
            #pragma unroll
            for (int r = 0; r < 8; ++r)
                Ot[(r + 8 * half) * 64 + nt * 16 + n] = acc[nt][r];
    }
#endif  // __HIP_DEVICE_COMPILE__
}

extern "C" void kernel_launch(void* const* d_in, const int* in_sizes, int n_in,
                              void* d_out, int out_size, void* d_ws, size_t ws_size,
                              hipStream_t stream) {
    const float* image = (const float*)d_in[0];   // [N,8,8] fp32
    const float* alpha = (const float*)d_in[1];   // [8,8]   fp32
    const float* basis = (const float*)d_in[2];   // [8,8,8,8] fp32
    float* out = (float*)d_out;

    const int n_blocks = in_sizes[0] / 64;        // N (2^20 in the reference)
    const int n_tiles  = n_blocks / 16;           // 16 blocks per wave-tile

    int blocks = 2048;                            // 16K waves for latency hiding
    if (blocks * 8 > n_tiles) blocks = (n_tiles + 7) / 8;
    if (blocks < 1) blocks = 1;

    idct8x8_wmma_f32_kernel<<<blocks, 256, 0, stream>>>(image, alpha, basis, out, n_tiles);
}